// Prediction_35579509080741
// MI455X (gfx1250) — compile-verified
//
#include <hip/hip_runtime.h>
#include <hip/hip_bf16.h>

#define BB    16
#define CC    80
#define HH    128
#define WW    128
#define HWSZ  (HH * WW)          // 16384
#define NBINS 2048               // float bits >> 20 (positive floats -> bin <= 2047)
#define CAP   8192               // per-batch candidate capacity
#define TOPK  100
#define THRESH 0.01f

// ---------- CDNA5 async global->LDS path (guarded; falls back to float4 copy) ----------
#if defined(__has_builtin)
#  if __has_builtin(__builtin_amdgcn_global_load_async_to_lds_b128) && \
      __has_builtin(__builtin_amdgcn_s_wait_asynccnt)
#    define USE_ASYNC_LDS 1
#  endif
#endif
#ifndef USE_ASYNC_LDS
#  define USE_ASYNC_LDS 0
#endif

#if USE_ASYNC_LDS
typedef int v4i __attribute__((ext_vector_type(4)));
#define AS1 __attribute__((address_space(1)))
#define AS3 __attribute__((address_space(3)))
static __device__ __forceinline__ void async_copy_b128(const void* gsrc, void* ldst) {
  __builtin_amdgcn_global_load_async_to_lds_b128(
      (AS1 v4i*)gsrc, (AS3 v4i*)ldst, 0, 0);
}
#endif

// Load 65 rows (8320 floats = 2080 float4) of one half-plane tile into LDS.
static __device__ __forceinline__ void load_tile(const float* __restrict__ plane,
                                                 float* tile, int t) {
#if USE_ASYNC_LDS
  for (int j = t; j < 2080; j += 256)
    async_copy_b128(plane + j * 4, (char*)tile + j * 16);
  __builtin_amdgcn_s_wait_asynccnt(0);
#else
  for (int j = t; j < 2080; j += 256)
    ((float4*)tile)[j] = ((const float4*)plane)[j];
#endif
}

// Peak test for element (y,x) of full plane; tile row index is ly.
static __device__ __forceinline__ bool is_peak(const float* tile, int ly, int x,
                                               int y, float v) {
  const float* row = tile + ly * WW + x;
  bool lf = x > 0, rt = x < WW - 1;
  bool p = true;
  if (lf) p = p && (v >= row[-1]);
  if (rt) p = p && (v >= row[1]);
  if (y > 0) {
    const float* ru = row - WW;
    p = p && (v >= ru[0]);
    if (lf) p = p && (v >= ru[-1]);
    if (rt) p = p && (v >= ru[1]);
  }
  if (y < HH - 1) {
    const float* rd = row + WW;
    p = p && (v >= rd[0]);
    if (lf) p = p && (v >= rd[-1]);
    if (rt) p = p && (v >= rd[1]);
  }
  return p;
}

// ---------------- Pass 1: peaks + per-batch value histogram ----------------
__global__ __launch_bounds__(256) void k_peaks_hist(const float* __restrict__ heat,
                                                    unsigned* __restrict__ ghist) {
  __shared__ __align__(16) float tile[65 * WW];
  __shared__ unsigned lhist[NBINS];
  int t = threadIdx.x;
  int bid = blockIdx.x;
  int half = bid & 1;
  int bc = bid >> 1;            // b*CC + c
  int b = bc / CC;
  int r0 = half * 64;
  int gstart = half ? 63 : 0;   // 65 rows loaded: [gstart, gstart+64]
  const float* plane = heat + (size_t)bc * HWSZ + (size_t)gstart * WW;

  for (int j = t; j < NBINS; j += 256) lhist[j] = 0u;
  load_tile(plane, tile, t);
  __syncthreads();

  int lyoff = r0 - gstart;      // 0 (top half) or 1 (bottom half)
  for (int k = 0; k < 32; ++k) {
    int e = k * 256 + t;        // 0..8191 within the 64-row region
    int yr = e >> 7;
    int x = e & (WW - 1);
    int y = r0 + yr;
    int ly = yr + lyoff;
    float v = tile[ly * WW + x];
    if (v > THRESH && is_peak(tile, ly, x, y, v))
      atomicAdd(&lhist[__float_as_uint(v) >> 20], 1u);
  }
  __syncthreads();

  unsigned* gh = ghist + (size_t)b * NBINS;
  for (int j = t; j < NBINS; j += 256) {
    unsigned v = lhist[j];
    if (v) atomicAdd(&gh[j], v);
  }
}

// ---------------- Pass 2: find per-batch cut bin (100th largest) ----------------
__global__ __launch_bounds__(256) void k_findcut(const unsigned* __restrict__ ghist,
                                                 unsigned* __restrict__ cut) {
  __shared__ unsigned part[256];
  int b = blockIdx.x, t = threadIdx.x;
  const unsigned* h = ghist + (size_t)b * NBINS;
  const int CHUNK = NBINS / 256;   // 8
  unsigned s = 0;
  for (int j = 0; j < CHUNK; ++j) s += h[t * CHUNK + j];
  part[t] = s;
  __syncthreads();
  if (t == 0) {
    unsigned acc = 0, cutBin = 0;
    for (int c = 255; c >= 0; --c) {
      if (acc + part[c] >= TOPK) {
        for (int j = CHUNK - 1; j >= 0; --j) {
          acc += h[c * CHUNK + j];
          if (acc >= TOPK) { cutBin = (unsigned)(c * CHUNK + j); break; }
        }
        break;
      }
      acc += part[c];
    }
    cut[b] = cutBin;   // 0 if fewer than TOPK candidates exist -> take all
  }
}

// ---------------- Pass 3: collect candidates in bins >= cut ----------------
__global__ __launch_bounds__(256) void k_collect(const float* __restrict__ heat,
                                                 const unsigned* __restrict__ cut,
                                                 unsigned* __restrict__ cnt,
                                                 unsigned long long* __restrict__ cand) {
  __shared__ __align__(16) float tile[65 * WW];
  int t = threadIdx.x;
  int bid = blockIdx.x;
  int half = bid & 1;
  int bc = bid >> 1;
  int b = bc / CC, c = bc % CC;
  int r0 = half * 64;
  int gstart = half ? 63 : 0;
  const float* plane = heat + (size_t)bc * HWSZ + (size_t)gstart * WW;
  unsigned cutBin = cut[b];

  load_tile(plane, tile, t);
  __syncthreads();

  int lyoff = r0 - gstart;
  for (int k = 0; k < 32; ++k) {
    int e = k * 256 + t;
    int yr = e >> 7;
    int x = e & (WW - 1);
    int y = r0 + yr;
    int ly = yr + lyoff;
    float v = tile[ly * WW + x];
    unsigned bits = __float_as_uint(v);
    if (v > THRESH && (bits >> 20) >= cutBin && is_peak(tile, ly, x, y, v)) {
      unsigned flat = (unsigned)(c * HWSZ + y * WW + x);
      unsigned pos = atomicAdd(&cnt[b], 1u);
      if (pos < CAP)
        cand[(size_t)b * CAP + pos] =
            ((unsigned long long)bits << 32) | (unsigned long long)(0xFFFFFFFFu - flat);
    }
  }
}

// ---------------- Pass 4: exact top-100 per batch + decode/gather/write ----------------
__global__ __launch_bounds__(256) void k_topk_out(const unsigned long long* __restrict__ cand,
                                                  const unsigned* __restrict__ cnt,
                                                  const float* __restrict__ off,
                                                  const float* __restrict__ wh,
                                                  float* __restrict__ out) {
  __shared__ unsigned long long red[256];
  __shared__ unsigned long long sel[TOPK];
  int b = blockIdx.x, t = threadIdx.x;
  unsigned M = cnt[b]; if (M > CAP) M = CAP;
  const unsigned long long* cb = cand + (size_t)b * CAP;

  unsigned long long keys[CAP / 256];   // 32 keys per thread in VGPRs
#pragma unroll
  for (int j = 0; j < CAP / 256; ++j) {
    unsigned i = (unsigned)t + (unsigned)j * 256u;
    keys[j] = (i < M) ? cb[i] : 0ull;
  }

  unsigned long long last = ~0ull;
  for (int k = 0; k < TOPK; ++k) {
    unsigned long long loc = 0ull;
#pragma unroll
    for (int j = 0; j < CAP / 256; ++j) {
      unsigned long long key = keys[j];
      if (key < last && key > loc) loc = key;
    }
    red[t] = loc;
    __syncthreads();
    for (int s2 = 128; s2 > 0; s2 >>= 1) {
      if (t < s2) { if (red[t + s2] > red[t]) red[t] = red[t + s2]; }
      __syncthreads();
    }
    unsigned long long best = red[0];
    if (t == 0) sel[k] = best;
    last = best;                 // keys are unique; 0 sentinel stays extracted forever
    __syncthreads();
  }

  if (t < TOPK) {
    unsigned long long key = sel[t];
    int o = b * TOPK + t;
    float* ids    = out;
    float* scores = out + BB * TOPK;
    float* bbox   = out + 2 * BB * TOPK;
    if (key == 0ull) {           // fewer than TOPK candidates, or sub-threshold slot
      ids[o] = -1.0f; scores[o] = -1.0f;
      bbox[4 * o + 0] = -4.0f; bbox[4 * o + 1] = -4.0f;
      bbox[4 * o + 2] = -4.0f; bbox[4 * o + 3] = -4.0f;
    } else {
      unsigned bits = (unsigned)(key >> 32);
      unsigned flat = 0xFFFFFFFFu - (unsigned)(key & 0xFFFFFFFFu);
      float v = __uint_as_float(bits);
      unsigned cls = flat / HWSZ;
      unsigned sp  = flat % HWSZ;
      float ysf = (float)(sp >> 7);
      float xsf = (float)(sp & (WW - 1));
      size_t obase = (size_t)b * 2 * HWSZ + sp;
      float ox = off[obase], oy = off[obase + HWSZ];
      float w_ = wh[obase],  h_ = wh[obase + HWSZ];
      float cx = xsf + ox, cy = ysf + oy;
      ids[o] = (float)cls;
      scores[o] = v;
      bbox[4 * o + 0] = (cx - w_ * 0.5f) * 4.0f;
      bbox[4 * o + 1] = (cy - h_ * 0.5f) * 4.0f;
      bbox[4 * o + 2] = (cx + w_ * 0.5f) * 4.0f;
      bbox[4 * o + 3] = (cy + h_ * 0.5f) * 4.0f;
    }
  }
}

extern "C" void kernel_launch(void* const* d_in, const int* in_sizes, int n_in,
                              void* d_out, int out_size, void* d_ws, size_t ws_size,
                              hipStream_t stream) {
  (void)in_sizes; (void)n_in; (void)out_size; (void)ws_size;
  const float* heat = (const float*)d_in[0];
  const float* off  = (const float*)d_in[1];
  const float* wh   = (const float*)d_in[2];
  float* out = (float*)d_out;

  char* ws = (char*)d_ws;
  unsigned* ghist = (unsigned*)ws;                           // 16*2048*4 = 131072 B
  unsigned* cut   = (unsigned*)(ws + 131072);                // 16*4
  unsigned* cnt   = (unsigned*)(ws + 131136);                // 16*4
  unsigned long long* cand = (unsigned long long*)(ws + 131200); // 16*8192*8 = 1 MB

  (void)hipMemsetAsync(d_ws, 0, 131200, stream);             // hist + cut + counters

  k_peaks_hist<<<BB * CC * 2, 256, 0, stream>>>(heat, ghist);
  k_findcut  <<<BB,           256, 0, stream>>>(ghist, cut);
  k_collect  <<<BB * CC * 2, 256, 0, stream>>>(heat, cut, cnt, cand);
  k_topk_out <<<BB,           256, 0, stream>>>(cand, cnt, off, wh, out);
}